// CIF_18150531793024
// MI455X (gfx1250) — compile-verified
//
#include <hip/hip_runtime.h>

// ---- problem constants (match reference) ----
#define BB      8
#define TT      8192
#define DD      256
#define TAPS    5
#define KR      2
#define MAXY    1024
#define BT      (BB*TT)
#define THRESH  0.9f

typedef float v2f __attribute__((ext_vector_type(2)));
typedef float v8f __attribute__((ext_vector_type(8)));

// ---------------------------------------------------------------------------
// K0: fold proj into conv -> effw16[16*256] (rows 5..15 zero), effb;
//     init fire-slot table to -1.
// conv_w layout (W,I,O): conv_w[(k*DD+din)*DD + dout]
// ---------------------------------------------------------------------------
__global__ void cif_init(const float* __restrict__ conv_w,
                         const float* __restrict__ conv_b,
                         const float* __restrict__ proj_w,
                         const float* __restrict__ proj_b,
                         float* __restrict__ effw16,
                         float* __restrict__ effb,
                         int* __restrict__ tfire) {
  int tid = blockIdx.x * blockDim.x + threadIdx.x;
  if (tid < 16 * DD) {
    const int n = tid >> 8;             // padded tap row
    float s = 0.f;
    if (n < TAPS) {
      const float* row = conv_w + (size_t)tid * DD;  // contiguous dout run
      for (int o = 0; o < DD; ++o) s += row[o] * proj_w[o];
    }
    effw16[tid] = s;                    // rows 5..15 -> 0
  }
  if (tid == 16 * DD) {
    float s = proj_b[0];
    for (int o = 0; o < DD; ++o) s += conv_b[o] * proj_w[o];
    effb[0] = s;
  }
  if (tid < BB * MAXY) tfire[tid] = -1;
}

// ---------------------------------------------------------------------------
// K1: per-tap dots via WMMA f32 16x16x4, two 16-row C tiles per wave.
// d5[k][row] = dot(e[row,:], effw[k,:])  for k=0..4  (row = b*T + t)
// A frag: lane l holds M = l&15; lanes<16 carry K={0,1}, lanes>=16 K={2,3}.
// B frag: N = l&15; same half-K lane-group split (ISA 7.12.2). Shared by
// both tiles -> halves B traffic, gives two independent wmma chains.
// ---------------------------------------------------------------------------
__global__ void __launch_bounds__(256) tapdot_wmma(const float* __restrict__ e,
                                                   const float* __restrict__ effw16,
                                                   float* __restrict__ d5) {
  const int wave = threadIdx.x >> 5;
  const int lane = threadIdx.x & 31;
  const int m    = lane & 15;
  const int grp  = lane >> 4;
  const int r0   = blockIdx.x * 256 + wave * 32;    // 32 time-rows per wave

  const float* arow0 = e + (size_t)(r0 + m) * DD;
  const float* arow1 = e + (size_t)(r0 + 16 + m) * DD;
  const float* brow  = effw16 + (size_t)m * DD;     // pre-zeroed cols 5..15

  v8f c0 = {}, c1 = {};
  for (int ko = 0; ko < DD / 4; ++ko) {
    const int dbase = ko * 4 + grp * 2;
    v2f b, a0, a1;
    b.x  = brow[dbase];   b.y  = brow[dbase + 1];
    a0.x = arow0[dbase];  a0.y = arow0[dbase + 1];
    a1.x = arow1[dbase];  a1.y = arow1[dbase + 1];
    // 8 args: (neg_a, A, neg_b, B, c_mod, C, reuse_a, reuse_b)
    c0 = __builtin_amdgcn_wmma_f32_16x16x4_f32(false, a0, false, b, (short)0,
                                               c0, false, false);
    c1 = __builtin_amdgcn_wmma_f32_16x16x4_f32(false, a1, false, b, (short)0,
                                               c1, false, false);
  }

  // C layout: lane grp0 vgpr r -> M=r ; grp1 -> M=8+r ; N = lane&15
  if (m < TAPS) {
    float* plane = d5 + (size_t)m * BT;
    const int rowbase = r0 + grp * 8;
#pragma unroll
    for (int r = 0; r < 8; ++r) {
      plane[rowbase + r]      = c0[r];
      plane[rowbase + 16 + r] = c1[r];
    }
  }
}

// ---------------------------------------------------------------------------
// K2: alpha[b,t] = sigmoid(sum_k d5[k][t+k-2] + effb) ; per-block partial sums.
// T/256 = 32 blocks per batch -> each block lies inside one batch.
// ---------------------------------------------------------------------------
__global__ void __launch_bounds__(256) alpha_stencil(const float* __restrict__ d5,
                                                     const float* __restrict__ effb,
                                                     float* __restrict__ alpha_out,
                                                     float* __restrict__ partials) {
  __shared__ float sdata[256];
  const int gid = blockIdx.x * 256 + threadIdx.x;
  const int b = gid / TT, t = gid % TT;
  float s = effb[0];
#pragma unroll
  for (int k = 0; k < TAPS; ++k) {
    const int tt = t + k - KR;
    if (tt >= 0 && tt < TT) s += d5[(size_t)k * BT + (size_t)b * TT + tt];
  }
  const float a = 1.0f / (1.0f + __expf(-s));
  alpha_out[gid] = a;

  sdata[threadIdx.x] = a;
  __syncthreads();
  for (int off = 128; off > 0; off >>= 1) {
    if (threadIdx.x < off) sdata[threadIdx.x] += sdata[threadIdx.x + off];
    __syncthreads();
  }
  if (threadIdx.x == 0) partials[blockIdx.x] = sdata[0];
}

// K3: deterministic finalize of per-batch alpha sums (32 partials per batch).
__global__ void sum_finalize(const float* __restrict__ partials,
                             float* __restrict__ asum) {
  const int tid = threadIdx.x;
  if (tid < BB) {
    float s = 0.f;
    for (int i = 0; i < 32; ++i) s += partials[tid * 32 + i];
    asum[tid] = s;
  }
}

// ---------------------------------------------------------------------------
// K4: scalar CIF scan. 8 batches = 8 SIMD lanes of one wave.
// Stores ONLY per-fire records (time, closing weight, opening weight);
// non-fire weights are recomputed in K5 as alpha[t]*scale (same FP expr).
// Faithful to reference ordering (residual updated first).
// ---------------------------------------------------------------------------
__global__ void cif_scan(const float* __restrict__ alpha,
                         const int* __restrict__ elens,
                         const int* __restrict__ ylens,
                         const float* __restrict__ asum,
                         int* __restrict__ tfire,
                         float* __restrict__ wclose,
                         float* __restrict__ wopen) {
  const int b = threadIdx.x;
  if (b >= BB) return;
  const float scale = (float)ylens[b] / asum[b];
  const int elen = elens[b];
  const float* ab = alpha + (size_t)b * TT;
  int*   fb = tfire  + b * MAXY;
  float* wc = wclose + b * MAXY;
  float* wo = wopen  + b * MAXY;

  float a_res = 0.f;
  int idx = 0;
  const int tmax = (elen < TT) ? elen : TT;   // trailing steps are inert
  for (int t = 0; t < tmax; ++t) {
    const float a = ab[t] * scale;
    const float accum = a + a_res;
    if (accum >= THRESH) {
      const float new_res = a - (1.0f - a_res);  // residual updated FIRST
      if (idx < MAXY) {
        fb[idx] = t;
        wc[idx] = 1.0f - new_res;                // closes segment idx at t
        wo[idx] = new_res;                       // opens segment idx+1 at t
      }
      ++idx;
      a_res = new_res;
    } else {
      a_res = accum;
    }
  }
}

// ---------------------------------------------------------------------------
// K5: out[b,y,:] = wopen[y-1]*e[t0,:]
//                + sum_{t in (t0,t1)} (alpha[t]*scale)*e[t,:]
//                + wclose[y]*e[t1,:]
// One block per (b,y); 256 threads = D channels (coalesced). Unfired -> 0.
// ---------------------------------------------------------------------------
__global__ void __launch_bounds__(256) segment_sum(const float* __restrict__ e,
                                                   const float* __restrict__ alpha,
                                                   const int* __restrict__ ylens,
                                                   const float* __restrict__ asum,
                                                   const int* __restrict__ tfire,
                                                   const float* __restrict__ wclose,
                                                   const float* __restrict__ wopen,
                                                   float* __restrict__ out) {
  const int y = blockIdx.x % MAXY;
  const int b = blockIdx.x / MAXY;
  const int d = threadIdx.x;
  float* orow = out + ((size_t)(b * MAXY + y)) * DD + d;

  const int t1 = tfire[b * MAXY + y];
  if (t1 < 0) { *orow = 0.f; return; }

  const float scale = (float)ylens[b] / asum[b];   // same expr as scan
  const float* eb = e + (size_t)b * TT * DD;
  const float* ab = alpha + (size_t)b * TT;

  float acc = 0.f;
  int tstart = 0;
  if (y > 0) {
    const int t0 = tfire[b * MAXY + y - 1];
    acc = wopen[b * MAXY + y - 1] * eb[(size_t)t0 * DD + d];
    tstart = t0 + 1;
  }
  for (int t = tstart; t < t1; ++t)
    acc += (ab[t] * scale) * eb[(size_t)t * DD + d];
  acc += wclose[b * MAXY + y] * eb[(size_t)t1 * DD + d];
  *orow = acc;
}

// ---------------------------------------------------------------------------
extern "C" void kernel_launch(void* const* d_in, const int* in_sizes, int n_in,
                              void* d_out, int out_size, void* d_ws, size_t ws_size,
                              hipStream_t stream) {
  const float* eouts  = (const float*)d_in[0];
  const int*   elens  = (const int*)d_in[1];
  const int*   ylens  = (const int*)d_in[2];
  const float* conv_w = (const float*)d_in[3];
  const float* conv_b = (const float*)d_in[4];
  const float* proj_w = (const float*)d_in[5];
  const float* proj_b = (const float*)d_in[6];

  float* out_fired = (float*)d_out;                       // [B, MAXY, D]
  float* out_alpha = out_fired + (size_t)BB * MAXY * DD;  // [B, T]

  // workspace carve (~1.4 MB)
  float* d5       = (float*)d_ws;                // 5*BT
  float* effw16   = d5 + (size_t)TAPS * BT;      // 16*256
  float* effb     = effw16 + 16 * DD;            // 1
  float* partials = effb + 1;                    // 256
  float* asum     = partials + 256;              // 8
  float* wclose   = asum + 8;                    // B*MAXY
  float* wopen    = wclose + BB * MAXY;          // B*MAXY
  int*   tfire    = (int*)(wopen + BB * MAXY);   // B*MAXY

  cif_init<<<32, 256, 0, stream>>>(conv_w, conv_b, proj_w, proj_b,
                                   effw16, effb, tfire);
  tapdot_wmma<<<BT / 256, 256, 0, stream>>>(eouts, effw16, d5);
  alpha_stencil<<<BT / 256, 256, 0, stream>>>(d5, effb, out_alpha, partials);
  sum_finalize<<<1, 32, 0, stream>>>(partials, asum);
  cif_scan<<<1, 32, 0, stream>>>(out_alpha, elens, ylens, asum,
                                 tfire, wclose, wopen);
  segment_sum<<<BB * MAXY, 256, 0, stream>>>(eouts, out_alpha, ylens, asum,
                                             tfire, wclose, wopen, out_fired);
}